// LogSinkhorn_14285061227131
// MI455X (gfx1250) — compile-verified
//
#include <hip/hip_runtime.h>
#include <math.h>

#define BATCH 4
#define NP 2048
#define DIM 256
#define EPSI 0.1f
#define INV_EPS 10.0f
#define THRESH_ 0.1f
#define TINYF 1e-8f
#define MAX_ITER 100
#define SZ (BATCH * NP * NP) /* 16,777,216 */

typedef __attribute__((ext_vector_type(2))) float v2f;
typedef __attribute__((ext_vector_type(8))) float v8f;

// ---------------------------------------------------------------------------
// init: u = v = 0, w_v = exp(v/eps) = 1, clear max/done
// ---------------------------------------------------------------------------
__global__ void k_init(float* u, float* v, float* w_v, unsigned* maxbits, unsigned* done) {
    int g = blockIdx.x * blockDim.x + threadIdx.x;
    if (g < BATCH * NP) { u[g] = 0.0f; v[g] = 0.0f; w_v[g] = 1.0f; }
    if (g == 0) { *maxbits = 0u; *done = 0u; }
}

// ---------------------------------------------------------------------------
// row norms: x2[b,i] = sum_d x^2, y2[b,j] = sum_d y^2 (one wave per row)
// ---------------------------------------------------------------------------
__global__ void k_norms(const float* __restrict__ x, const float* __restrict__ y,
                        float* __restrict__ x2, float* __restrict__ y2) {
    int wid = (blockIdx.x * blockDim.x + threadIdx.x) >> 5;
    int lane = threadIdx.x & 31;
    const float* src; float* dst; int row;
    if (wid < BATCH * NP) { src = x; dst = x2; row = wid; }
    else                  { src = y; dst = y2; row = wid - BATCH * NP; }
    const float* p = src + (size_t)row * DIM;
    float s = 0.0f;
    for (int t = lane; t < DIM; t += 32) { float v = p[t]; s += v * v; }
    for (int o = 16; o; o >>= 1) s += __shfl_xor(s, o);
    if (lane == 0) dst[row] = s;
}

// ---------------------------------------------------------------------------
// WMMA GEMM: C[b,i,j] = x2 + y2 - 2 * dot(x_i, y_j); track global max.
// One wave computes a 32x32 tile = 2x2 fragments of 16x16, K stepped by 4
// using V_WMMA_F32_16X16X4_F32 (f32 A/B/C, wave32 layouts per CDNA5 ISA).
// ---------------------------------------------------------------------------
__global__ void k_gemm(const float* __restrict__ x, const float* __restrict__ y,
                       const float* __restrict__ x2, const float* __restrict__ y2,
                       float* __restrict__ Cout, unsigned* __restrict__ maxbits) {
    int wid  = (blockIdx.x * blockDim.x + threadIdx.x) >> 5;   // 16384 waves
    int lane = threadIdx.x & 31;
    int hl   = lane >> 4;        // half-wave select (K pair)
    int lr   = lane & 15;        // row/col within fragment
    int b  = wid >> 12;          // 4096 tiles per batch (64x64 of 32x32)
    int t  = wid & 4095;
    int m0 = (t >> 6) << 5;
    int n0 = (t & 63) << 5;

    const float* xb = x + (size_t)b * NP * DIM;
    const float* yb = y + (size_t)b * NP * DIM;
    // A fragment: lanes 0-15 hold K={kk,kk+1}, lanes 16-31 hold K={kk+2,kk+3}
    const float* ax0 = xb + (size_t)(m0 + lr) * DIM + hl * 2;
    const float* ax1 = ax0 + 16 * DIM;
    // B = y^T fragment mirrors A: B[k][n] = y[n][k]
    const float* by0 = yb + (size_t)(n0 + lr) * DIM + hl * 2;
    const float* by1 = by0 + 16 * DIM;

    v8f acc[2][2];
#pragma unroll
    for (int mi = 0; mi < 2; mi++)
#pragma unroll
        for (int ni = 0; ni < 2; ni++)
#pragma unroll
            for (int e = 0; e < 8; e++) acc[mi][ni][e] = 0.0f;

    for (int kk = 0; kk < DIM; kk += 4) {
        v2f a0 = *(const v2f*)(ax0 + kk);
        v2f a1 = *(const v2f*)(ax1 + kk);
        v2f b0 = *(const v2f*)(by0 + kk);
        v2f b1 = *(const v2f*)(by1 + kk);
        acc[0][0] = __builtin_amdgcn_wmma_f32_16x16x4_f32(false, a0, false, b0, (short)0, acc[0][0], false, false);
        acc[0][1] = __builtin_amdgcn_wmma_f32_16x16x4_f32(false, a0, false, b1, (short)0, acc[0][1], false, false);
        acc[1][0] = __builtin_amdgcn_wmma_f32_16x16x4_f32(false, a1, false, b0, (short)0, acc[1][0], false, false);
        acc[1][1] = __builtin_amdgcn_wmma_f32_16x16x4_f32(false, a1, false, b1, (short)0, acc[1][1], false, false);
    }

    // Epilogue: C = x2 + y2 - 2*acc ; fragment layout: VGPR g -> M = g + 8*hl
    float* Cb = Cout + (size_t)b * NP * NP;
    const float* x2b = x2 + b * NP;
    const float* y2b = y2 + b * NP;
    float lmax = 0.0f;  // C >= 0 mathematically; clamp guards fp cancellation
#pragma unroll
    for (int mi = 0; mi < 2; mi++) {
#pragma unroll
        for (int ni = 0; ni < 2; ni++) {
            int col = n0 + ni * 16 + lr;
            float y2v = y2b[col];
#pragma unroll
            for (int g = 0; g < 8; g++) {
                int row = m0 + mi * 16 + g + hl * 8;
                float val = x2b[row] + y2v - 2.0f * acc[mi][ni][g];
                Cb[(size_t)row * NP + col] = val;
                lmax = fmaxf(lmax, val);
            }
        }
    }
    for (int o = 16; o; o >>= 1) lmax = fmaxf(lmax, __shfl_xor(lmax, o));
    if (lane == 0) atomicMax(maxbits, __float_as_uint(lmax)); // order-independent
}

__global__ void k_prep(const unsigned* maxbits, float* inv_max) {
    *inv_max = 1.0f / __uint_as_float(*maxbits);
}

// ---------------------------------------------------------------------------
// normalize C in place; K = exp(-C/eps) (Gibbs kernel, reused 200x from L2)
// ---------------------------------------------------------------------------
__global__ void k_expC(float* __restrict__ C, float* __restrict__ K,
                       const float* __restrict__ inv_max) {
    float im = *inv_max;
    unsigned base = blockIdx.x * 1024u + threadIdx.x;
#pragma unroll
    for (int r = 0; r < 4; r++) {
        unsigned idx = base + r * 256u;
        float cn = C[idx] * im;
        C[idx] = cn;
        K[idx] = __expf(-cn * INV_EPS);
    }
}

// ---------------------------------------------------------------------------
// u-update: r_i = sum_j K[b,i,j] * w_v[j]  (one wave per row, coalesced)
// u_new = eps*(log_mu - log(TINY + e^{u/eps} * r)) + u
// ---------------------------------------------------------------------------
__global__ void k_row(const float* __restrict__ K, float* __restrict__ u,
                      float* __restrict__ w_u, const float* __restrict__ w_v,
                      float* __restrict__ delta, const unsigned* __restrict__ done) {
    if (*done) return;
    int wid  = (blockIdx.x * blockDim.x + threadIdx.x) >> 5; // 0..8191
    int lane = threadIdx.x & 31;
    int b = wid >> 11, i = wid & 2047;
    const float* Kr = K + (size_t)(b * NP + i) * NP;
    const float* wv = w_v + b * NP;
    float s = 0.0f;
    for (int j = lane; j < NP; j += 32) s += Kr[j] * wv[j];
    for (int o = 16; o; o >>= 1) s += __shfl_xor(s, o);
    if (lane == 0) {
        float log_mu = logf(1.0f / NP + TINYF);
        float uo = u[b * NP + i];
        float r  = __expf(uo * INV_EPS) * s;
        float un = EPSI * (log_mu - logf(TINYF + r)) + uo;
        u[b * NP + i]     = un;
        w_u[b * NP + i]   = __expf(un * INV_EPS);
        delta[b * NP + i] = fabsf(un - uo);
    }
}

// ---------------------------------------------------------------------------
// v-update part 1: column partial sums over 256-row chunks (coalesced rows)
// ---------------------------------------------------------------------------
__global__ void k_colpart(const float* __restrict__ K, const float* __restrict__ w_u,
                          float* __restrict__ colpart, const unsigned* __restrict__ done) {
    if (*done) return;
    int b = blockIdx.z, rc = blockIdx.y;
    int col = blockIdx.x * 256 + threadIdx.x;
    const float* Kb = K + (size_t)b * NP * NP;
    const float* wu = w_u + b * NP;
    float acc = 0.0f;
    int i0 = rc * 256, i1 = i0 + 256;
    for (int i = i0; i < i1; i++) {
        if (i + 8 < i1)  // CDNA5 global_prefetch_b8: pull next rows into cache
            __builtin_prefetch((const void*)&Kb[(size_t)(i + 8) * NP + col], 0, 1);
        acc += Kb[(size_t)i * NP + col] * wu[i];
    }
    colpart[(b * 8 + rc) * NP + col] = acc;
}

// v-update part 2: combine 8 partials (fixed order), apply log update
__global__ void k_colfin(const float* __restrict__ colpart, float* __restrict__ v,
                         float* __restrict__ w_v, const unsigned* __restrict__ done) {
    if (*done) return;
    int g = blockIdx.x * 256 + threadIdx.x; // 0..8191
    int b = g >> 11, col = g & 2047;
    float c = 0.0f;
#pragma unroll
    for (int rc = 0; rc < 8; rc++) c += colpart[(b * 8 + rc) * NP + col];
    float log_nu = logf(1.0f / NP + TINYF);
    float vo = v[g];
    float s  = __expf(vo * INV_EPS) * c;
    float vn = EPSI * (log_nu - logf(TINYF + s)) + vo;
    v[g]   = vn;
    w_v[g] = __expf(vn * INV_EPS);
}

// err = max_b sum_i |du|; done |= err < thresh  (fixed-order LDS reduction)
__global__ void k_done(const float* __restrict__ delta, unsigned* __restrict__ done) {
    if (*done) return;
    __shared__ float sh[256];
    int t = threadIdx.x;
    float err = 0.0f;
    for (int b = 0; b < BATCH; b++) {
        float s = 0.0f;
        for (int i = t; i < NP; i += 256) s += delta[b * NP + i];
        sh[t] = s; __syncthreads();
        for (int o = 128; o; o >>= 1) { if (t < o) sh[t] += sh[t + o]; __syncthreads(); }
        if (t == 0) err = fmaxf(err, sh[0]);
        __syncthreads();
    }
    if (t == 0 && err < THRESH_) *done = 1u;
}

// ---------------------------------------------------------------------------
// pi = K * e^{u/eps} * e^{v/eps}; partial cost sums per block
// ---------------------------------------------------------------------------
__global__ void k_pi(const float* __restrict__ K, const float* __restrict__ Cn,
                     const float* __restrict__ w_u, const float* __restrict__ w_v,
                     float* __restrict__ pi_out, float* __restrict__ pipart) {
    __shared__ float sh[256];
    unsigned base = blockIdx.x * 1024u + threadIdx.x;
    float local = 0.0f;
#pragma unroll
    for (int r = 0; r < 4; r++) {
        unsigned idx = base + r * 256u;
        unsigned b = idx >> 22;            // / (NP*NP)
        unsigned rem = idx & (NP * NP - 1);
        unsigned i = rem >> 11, j = rem & 2047u;
        float p = K[idx] * w_u[b * NP + i] * w_v[b * NP + j];
        pi_out[idx] = p;
        local += p * Cn[idx];
    }
    sh[threadIdx.x] = local; __syncthreads();
    for (int o = 128; o; o >>= 1) { if (threadIdx.x < (unsigned)o) sh[threadIdx.x] += sh[threadIdx.x + o]; __syncthreads(); }
    if (threadIdx.x == 0) pipart[blockIdx.x] = sh[0];
}

__global__ void k_cost(const float* __restrict__ pipart, float* __restrict__ cost) {
    __shared__ float sh[256];
    int t = threadIdx.x;
    for (int b = 0; b < BATCH; b++) {
        float s = 0.0f;
        for (int p = t; p < 4096; p += 256) s += pipart[b * 4096 + p];
        sh[t] = s; __syncthreads();
        for (int o = 128; o; o >>= 1) { if (t < o) sh[t] += sh[t + o]; __syncthreads(); }
        if (t == 0) cost[b] = sh[0];
        __syncthreads();
    }
}

// ---------------------------------------------------------------------------
extern "C" void kernel_launch(void* const* d_in, const int* in_sizes, int n_in,
                              void* d_out, int out_size, void* d_ws, size_t ws_size,
                              hipStream_t stream) {
    const float* x = (const float*)d_in[0];
    const float* y = (const float*)d_in[1];
    float* out = (float*)d_out;
    float* ws  = (float*)d_ws;

    // workspace layout (floats)
    float* K       = ws;                     // SZ
    float* u       = ws + SZ;                // 8192
    float* v       = u + BATCH * NP;
    float* w_u     = v + BATCH * NP;
    float* w_v     = w_u + BATCH * NP;
    float* delta   = w_v + BATCH * NP;
    float* x2      = delta + BATCH * NP;
    float* y2      = x2 + BATCH * NP;
    float* colpart = y2 + BATCH * NP;        // 4*8*2048 = 65536
    float* pipart  = colpart + 65536;        // 16384
    float* scal    = pipart + 16384;
    unsigned* maxbits = (unsigned*)scal;
    unsigned* done    = (unsigned*)(scal + 1);
    float*    inv_max = scal + 2;

    float* cost   = out;            // (4,)
    float* pi_out = out + 4;        // (4,2048,2048)
    float* Cn     = out + 4 + SZ;   // (4,2048,2048)

    k_init <<<32,    256, 0, stream>>>(u, v, w_v, maxbits, done);
    k_norms<<<2048,  256, 0, stream>>>(x, y, x2, y2);
    k_gemm <<<2048,  256, 0, stream>>>(x, y, x2, y2, Cn, maxbits);
    k_prep <<<1,     1,   0, stream>>>(maxbits, inv_max);
    k_expC <<<16384, 256, 0, stream>>>(Cn, K, inv_max);

    for (int it = 0; it < MAX_ITER; it++) {
        k_row    <<<1024, 256, 0, stream>>>(K, u, w_u, w_v, delta, done);
        k_colpart<<<dim3(8, 8, BATCH), 256, 0, stream>>>(K, w_u, colpart, done);
        k_colfin <<<32,   256, 0, stream>>>(colpart, v, w_v, done);
        k_done   <<<1,    256, 0, stream>>>(delta, done);
    }

    k_pi  <<<16384, 256, 0, stream>>>(K, Cn, w_u, w_v, pi_out, pipart);
    k_cost<<<1,     256, 0, stream>>>(pipart, cost);
}